// NFTMRule110_16458314678599
// MI455X (gfx1250) — compile-verified
//
#include <hip/hip_runtime.h>

// ---------------------------------------------------------------------------
// NFTM "Rule110" CA:  out[0] = f0 ; out[t] = CA_rule(binarize(out[t-1]))
// rule = 8-bit LUT from the 3->16->1 controller (logit > 0 per neighborhood).
// Memory-bound: 512 MB of output stores -> bit-packed CA in LDS, NT b128 stores.
// ---------------------------------------------------------------------------

#define HIDDEN     16
#define B_DIM      32
#define N_DIM      65536
#define TILE_WORDS 128                 // 4096 cells per tile
#define TILE_CELLS (TILE_WORDS * 32)
#define WIN_WORDS  (TILE_WORDS + 4)    // +64-bit halo each side (covers 63 steps)
#define BLOCK      256

typedef __attribute__((ext_vector_type(2))) float v2f;
typedef __attribute__((ext_vector_type(4))) float v4f;
typedef __attribute__((ext_vector_type(8))) float v8f;

__device__ __forceinline__ unsigned ballot32(bool p) {
#if __has_builtin(__builtin_amdgcn_ballot_w32)
  return __builtin_amdgcn_ballot_w32(p);
#else
  return (unsigned)__ballot((int)p);
#endif
}

// ---------------------------------------------------------------------------
// Kernel 1: build the 8-entry rule LUT.  Layer 1 (8x3 @ 3x16, fp32) via
// V_WMMA_F32_16X16X4_F32 (exact fp32 -> same thresholds as reference).
// One wave of 32 threads; EXEC all-ones at the WMMA as required.
// ---------------------------------------------------------------------------
__global__ void rule_lut_kernel(const float* __restrict__ W1,  // (3,16) row-major
                                const float* __restrict__ b1,  // (16,)
                                const float* __restrict__ W2,  // (16,1)
                                const float* __restrict__ b2,  // (1,)
                                unsigned* __restrict__ ruleOut) {
  const int lane  = threadIdx.x;   // 0..31
  const int n     = lane & 15;     // column / hidden index
  const int khalf = lane >> 4;     // 0: K=0,1   1: K=2,3
  __shared__ float acc[8][16];
  float logit = 0.0f;

#if __has_builtin(__builtin_amdgcn_wmma_f32_16x16x4_f32)
  // A: 16x4, rows m = neighborhood index (rows 8..15 zero), K: 0=left,1=center,2=right,3=pad
  // B: 4x16 = W1 padded with zero K=3 row.
  v2f a, b;
  for (int v = 0; v < 2; ++v) {
    const int k = 2 * khalf + v;
    float av = 0.0f, bv = 0.0f;
    if (k < 3) {
      const int m = n;                       // A-row index for this lane
      if (m < 8) av = (float)((m >> (2 - k)) & 1);   // n = (l<<2)|(c<<1)|r
      bv = W1[k * 16 + n];
    }
    a[v] = av;
    b[v] = bv;
  }
  v8f c = {0.f, 0.f, 0.f, 0.f, 0.f, 0.f, 0.f, 0.f};
  v8f d = __builtin_amdgcn_wmma_f32_16x16x4_f32(false, a, false, b, (short)0, c,
                                                false, false);
  // lanes 0..15: d[r] = raw hidden pre-activation for neighborhood m=r, unit n
  if (khalf == 0) {
    const float bn = b1[n], wn = W2[n];
    for (int r = 0; r < 8; ++r) {
      float h = d[r] + bn;
      h = h > 0.0f ? h : 0.0f;
      acc[r][n] = h * wn;
    }
  }
  __syncthreads();
  if (lane < 8) {
    float s = b2[0];
    for (int j = 0; j < HIDDEN; ++j) s += acc[lane][j];
    logit = s;
  }
#else
  (void)acc;
  if (lane < 8) {
    const float l = (float)((lane >> 2) & 1);
    const float cc = (float)((lane >> 1) & 1);
    const float rr = (float)(lane & 1);
    float s = b2[0];
    for (int j = 0; j < HIDDEN; ++j) {
      float h = l * W1[0 * 16 + j] + cc * W1[1 * 16 + j] + rr * W1[2 * 16 + j] + b1[j];
      h = h > 0.0f ? h : 0.0f;
      s += h * W2[j];
    }
    logit = s;
  }
#endif
  const unsigned mask = ballot32((lane < 8) && (logit > 0.0f));
  if (lane == 0) ruleOut[0] = mask & 0xFFu;
}

// ---------------------------------------------------------------------------
// Kernel 2: bit-packed CA over a 4096-cell tile with 64-bit halos; all T-1
// steps in LDS (1 KB double buffer, 1 barrier/step); float expansion streamed
// with non-temporal 128-bit stores.
// ---------------------------------------------------------------------------
__global__ __launch_bounds__(BLOCK)
void nftm_ca_kernel(const float* __restrict__ f0,
                    const unsigned* __restrict__ rulePtr,
                    float* __restrict__ out, int T) {
  const int tile = blockIdx.x;
  const int b    = blockIdx.y;
  const int tid  = threadIdx.x;
  const int lane = tid & 31;
  const int wave = tid >> 5;

  __shared__ unsigned buf[2][WIN_WORDS + 2];   // index 0 / WIN_WORDS+1 = zero guards

  if (tid == 0) {
    buf[0][0] = 0u; buf[0][WIN_WORDS + 1] = 0u;
    buf[1][0] = 0u; buf[1][WIN_WORDS + 1] = 0u;
  }

  const int word0 = tile * TILE_WORDS - 2;     // window start (row word index)
  const float* frow = f0 + (size_t)b * N_DIM;

  // Binarize window into buf[0]: one wave32 ballot packs 32 cells per word.
  for (int j = wave; j < WIN_WORDS; j += (BLOCK / 32)) {
    const int rw   = word0 + j;
    const int cell = rw * 32 + lane;
    float v = 0.0f;
    if (cell >= 0 && cell < N_DIM) v = frow[cell];
    const unsigned m = ballot32(v > 0.5f);
    if (lane == 0) buf[0][1 + j] = m;
  }

  // t = 0: copy f0 tile verbatim (non-temporal, 16B stores).
  {
    const int base = tile * TILE_CELLS + tid * 16;
    float* orow = out + (size_t)b * N_DIM;
    const v4f* src = (const v4f*)(frow + base);
    v4f* dst = (v4f*)(orow + base);
    for (int q = 0; q < 4; ++q) {
      v4f val = src[q];
      __builtin_nontemporal_store(val, dst + q);
    }
  }

  const unsigned ruleMask = rulePtr[0];
  const unsigned r0 = (ruleMask &   1u) ? 0xFFFFFFFFu : 0u;
  const unsigned r1 = (ruleMask &   2u) ? 0xFFFFFFFFu : 0u;
  const unsigned r2 = (ruleMask &   4u) ? 0xFFFFFFFFu : 0u;
  const unsigned r3 = (ruleMask &   8u) ? 0xFFFFFFFFu : 0u;
  const unsigned r4 = (ruleMask &  16u) ? 0xFFFFFFFFu : 0u;
  const unsigned r5 = (ruleMask &  32u) ? 0xFFFFFFFFu : 0u;
  const unsigned r6 = (ruleMask &  64u) ? 0xFFFFFFFFu : 0u;
  const unsigned r7 = (ruleMask & 128u) ? 0xFFFFFFFFu : 0u;

  __syncthreads();

  for (int t = 1; t < T; ++t) {
    const unsigned* src = buf[(t - 1) & 1];
    unsigned* dst = buf[t & 1];
    if (tid < WIN_WORDS) {
      const int j = tid + 1;
      const unsigned C  = src[j];
      const unsigned Lw = src[j - 1];
      const unsigned Rw = src[j + 1];
      const unsigned L  = (C << 1) | (Lw >> 31);   // bit i = cell i-1
      const unsigned R  = (C >> 1) | (Rw << 31);   // bit i = cell i+1
      const unsigned nL = ~L, nC = ~C, nR = ~R;
      const unsigned w =
          (r0 & nL & nC & nR) | (r1 & nL & nC &  R) |
          (r2 & nL &  C & nR) | (r3 & nL &  C &  R) |
          (r4 &  L & nC & nR) | (r5 &  L & nC &  R) |
          (r6 &  L &  C & nR) | (r7 &  L &  C &  R);
      dst[j] = w;
    }
    __syncthreads();   // single barrier per step (double-buffered)

    // Expand this step's tile region: 16 cells/thread -> 4 NT b128 stores.
    const unsigned bits = dst[3 + (tid >> 1)] >> ((tid & 1) * 16);
    float* op = out + ((size_t)t * B_DIM + b) * N_DIM + tile * TILE_CELLS + tid * 16;
    for (int q = 0; q < 4; ++q) {
      v4f val;
      val[0] = (bits >> (q * 4 + 0)) & 1u ? 1.0f : 0.0f;
      val[1] = (bits >> (q * 4 + 1)) & 1u ? 1.0f : 0.0f;
      val[2] = (bits >> (q * 4 + 2)) & 1u ? 1.0f : 0.0f;
      val[3] = (bits >> (q * 4 + 3)) & 1u ? 1.0f : 0.0f;
      __builtin_nontemporal_store(val, (v4f*)op + q);
    }
  }
}

// ---------------------------------------------------------------------------
extern "C" void kernel_launch(void* const* d_in, const int* in_sizes, int n_in,
                              void* d_out, int out_size, void* d_ws, size_t ws_size,
                              hipStream_t stream) {
  const float* f0 = (const float*)d_in[0];
  const float* W1 = (const float*)d_in[1];
  const float* b1 = (const float*)d_in[2];
  const float* W2 = (const float*)d_in[3];
  const float* b2 = (const float*)d_in[4];
  float* out = (float*)d_out;
  unsigned* rule = (unsigned*)d_ws;

  const int T = out_size / (B_DIM * N_DIM);   // 64

  hipLaunchKernelGGL(rule_lut_kernel, dim3(1), dim3(32), 0, stream,
                     W1, b1, W2, b2, rule);

  dim3 grid(N_DIM / TILE_CELLS, B_DIM);       // 16 x 32 = 512 blocks
  hipLaunchKernelGGL(nftm_ca_kernel, grid, dim3(BLOCK), 0, stream,
                     f0, rule, out, T);
}